// flex_conv_op_66726611910774
// MI455X (gfx1250) — compile-verified
//
#include <hip/hip_runtime.h>

typedef float v2f __attribute__((ext_vector_type(2)));
typedef float v4f __attribute__((ext_vector_type(4)));
typedef float v8f __attribute__((ext_vector_type(8)));

#define BS 16
#define IC 16
#define OC 32
#define HH 64
#define WW 512
#define HP 62
#define WP 510
#define GROUPS_W 4     // 4 groups of 8 tiles -> 32 tiles of 16 pixels = 512 >= 510
#define GCOLS 130      // 128 pixels + 2 halo columns staged per group
#define XSTRIDE 132    // padded LDS row stride (floats)

__global__ __launch_bounds__(256) void flexconv_wmma_f32_lds(
    const float* __restrict__ x,      // (16,16,64,512)
    const float* __restrict__ kern,   // (32,16,4)
    const float* __restrict__ pts,    // (16,3,64,512)
    float* __restrict__ out)          // (16,32,62,510)
{
    __shared__ float xs[IC * 3 * XSTRIDE];   // features: ch x row x col
    __shared__ float ps[3 * 3 * XSTRIDE];    // points:   dim x row x col
    __shared__ float kw[OC * IC * 4];        // kernel weights

    const int tid  = threadIdx.x;
    const int lane = tid & 31;
    const int wave = tid >> 5;

    const int grp = blockIdx.x % GROUPS_W;
    const int h   = (blockIdx.x / GROUPS_W) % HP;
    const int b   = blockIdx.x / (GROUPS_W * HP);

    const int w0g = grp * 128;            // first pixel column of this group
    const int n    = lane & 15;           // pixel within this wave's 16-wide tile
    const int half = lane >> 4;           // 0: K%4 in {0,1}; 1: K%4 in {2,3}
    const int w0   = w0g + wave * 16;     // this wave's first pixel column

    // ---------------- cooperative LDS staging (coalesced) ----------------
    {   // kernel weights: 2048 floats = 512 v4f, 2 per thread
        const v4f* src = (const v4f*)kern;
        v4f* dst = (v4f*)kw;
        dst[tid * 2]     = src[tid * 2];
        dst[tid * 2 + 1] = src[tid * 2 + 1];
    }
    {   // x halo: 16ch x 3rows x 130cols
        const float* xb = x + (size_t)b * IC * HH * WW;
        for (int idx = tid; idx < IC * 3 * GCOLS; idx += 256) {
            int ch  = idx / (3 * GCOLS);
            int rem = idx - ch * (3 * GCOLS);
            int row = rem / GCOLS;
            int col = rem - row * GCOLS;
            int gc  = w0g + col;
            if (gc > WW - 1) gc = WW - 1;
            xs[(ch * 3 + row) * XSTRIDE + col] =
                xb[(size_t)ch * HH * WW + (size_t)(h + row) * WW + gc];
        }
    }
    {   // points halo: 3dim x 3rows x 130cols
        const float* pb = pts + (size_t)b * 3 * HH * WW;
        for (int idx = tid; idx < 3 * 3 * GCOLS; idx += 256) {
            int d   = idx / (3 * GCOLS);
            int rem = idx - d * (3 * GCOLS);
            int row = rem / GCOLS;
            int col = rem - row * GCOLS;
            int gc  = w0g + col;
            if (gc > WW - 1) gc = WW - 1;
            ps[(d * 3 + row) * XSTRIDE + col] =
                pb[(size_t)d * HH * WW + (size_t)(h + row) * WW + gc];
        }
    }
    __syncthreads();

    // local (LDS) pixel column, clamped identically to the old global clamp
    int wl = wave * 16 + n;
    {
        int wmax = (WP - 1) - w0g;        // clamp so window cols stay staged & real
        if (wl > wmax) wl = wmax;
    }

    // ---- point windows from LDS: low lanes d=0,1; high lanes d=2 (+const 1) ----
    float pA[9], pB[9], ctrA, ctrB;
    if (half == 0) {
#pragma unroll
        for (int k = 0; k < 3; ++k)
#pragma unroll
            for (int f = 0; f < 3; ++f) {
                pA[k * 3 + f] = ps[(0 * 3 + k) * XSTRIDE + wl + f];
                pB[k * 3 + f] = ps[(1 * 3 + k) * XSTRIDE + wl + f];
            }
        ctrA = ps[(0 * 3 + 1) * XSTRIDE + wl + 1];
        ctrB = ps[(1 * 3 + 1) * XSTRIDE + wl + 1];
    } else {
#pragma unroll
        for (int k = 0; k < 3; ++k)
#pragma unroll
            for (int f = 0; f < 3; ++f) {
                pA[k * 3 + f] = ps[(2 * 3 + k) * XSTRIDE + wl + f];
                pB[k * 3 + f] = 0.0f;                  // d=3 slot -> box sum
            }
        ctrA = ps[(2 * 3 + 1) * XSTRIDE + wl + 1];
        ctrB = 1.0f;
    }

    // ---- A matrices (kernel weights) from LDS, register-resident ----
    v2f a0[IC], a1[IC];
#pragma unroll
    for (int i = 0; i < IC; ++i) {
        a0[i] = *(const v2f*)(kw + ((n     ) * IC + i) * 4 + 2 * half);
        a1[i] = *(const v2f*)(kw + ((n + 16) * IC + i) * 4 + 2 * half);
    }

    v8f c0 = {};
    v8f c1 = {};

    // ---- per-channel window reduction (LDS reads) + WMMA over K=64 ----
#pragma unroll
    for (int i = 0; i < IC; ++i) {
        const float* xi = &xs[i * 3 * XSTRIDE];
        float box = 0.0f, dA = 0.0f, dB = 0.0f;
#pragma unroll
        for (int k = 0; k < 3; ++k)
#pragma unroll
            for (int f = 0; f < 3; ++f) {
                float xv = xi[k * XSTRIDE + wl + f];
                box += xv;
                dA = fmaf(pA[k * 3 + f], xv, dA);
                dB = fmaf(pB[k * 3 + f], xv, dB);
            }
        v2f bvec;
        bvec.x = fmaf(ctrA, box, -dA);   // K%4 = 0 (low) / 2 (high)
        bvec.y = fmaf(ctrB, box, -dB);   // K%4 = 1 (low) / 3 (high)

        c0 = __builtin_amdgcn_wmma_f32_16x16x4_f32(
                false, a0[i], false, bvec, (short)0, c0, false, false);
        c1 = __builtin_amdgcn_wmma_f32_16x16x4_f32(
                false, a1[i], false, bvec, (short)0, c1, false, false);
    }

    // ---- store: C/D layout: VGPR v -> M = v + 8*half, N = lane&15 ----
    if (w0 + n < WP) {
        float* ob = out + ((size_t)(b * OC) * HP + h) * WP + (w0 + n);
#pragma unroll
        for (int v = 0; v < 8; ++v) {
            const int o0 = v + 8 * half;
            ob[(size_t)(o0     ) * HP * WP] = c0[v];
            ob[(size_t)(o0 + 16) * HP * WP] = c1[v];
        }
    }
}

extern "C" void kernel_launch(void* const* d_in, const int* in_sizes, int n_in,
                              void* d_out, int out_size, void* d_ws, size_t ws_size,
                              hipStream_t stream) {
    const float* x    = (const float*)d_in[0];   // (16,16,64,512)
    const float* kern = (const float*)d_in[1];   // (32,16,4)
    const float* pts  = (const float*)d_in[2];   // (16,3,64,512)
    float* out = (float*)d_out;

    const int blocks = BS * HP * GROUPS_W;   // 16*62*4 = 3968 blocks of 8 waves
    flexconv_wmma_f32_lds<<<blocks, 256, 0, stream>>>(x, kern, pts, out);
}